// Clustering_79980880986445
// MI455X (gfx1250) — compile-verified
//
#include <hip/hip_runtime.h>

// Capsule "Clustering" layer on MI455X (gfx1250).
// Routing is algebraically degenerate (softmax over the summed axis), so:
//   out[b, nout] = sum_k ( sum_i u_vecs[b,i,k] ) * W[k, nout]
// Stage 1: streaming reduction of u_vecs over I (64 MB read = the roofline
//          cost, ~2.75 us at 23.3 TB/s) using b128 vector loads.
// Stage 2: [32x512]x[512x512] fp32 GEMM via V_WMMA_F32_16X16X4_F32.

typedef __attribute__((ext_vector_type(2))) float v2f;
typedef __attribute__((ext_vector_type(4))) float v4f;
typedef __attribute__((ext_vector_type(8))) float v8f;

#define B_SZ   32
#define I_SZ   1024
#define DIN    512
#define NOUT   512            // NUM_CAPSULE * DIM_CAPSULE = 32*16
#define ICH    16             // I-split factor for the reduction
#define ICHUNK (I_SZ / ICH)   // 64

// ---------------------------------------------------------------------------
// Stage 1a: partial reduction over I.  grid (B=32, ICH=16), 128 threads.
// Each lane owns 4 consecutive k -> one global_load_b128 per row per lane;
// a 128-lane block covers a full 512-float row (2 KB) per iteration.
// ---------------------------------------------------------------------------
__global__ void caps_reduce_partial(const float* __restrict__ u,
                                    float* __restrict__ part) {
  const int t   = threadIdx.x;   // 0..127  -> k4 = t*4
  const int b   = blockIdx.x;    // 0..31
  const int ich = blockIdx.y;    // 0..15
  const v4f* p = (const v4f*)(u + ((size_t)b * I_SZ + (size_t)ich * ICHUNK) * DIN) + t;
  v4f s = {};
#pragma unroll 8
  for (int i = 0; i < ICHUNK; ++i) s += p[(size_t)i * (DIN / 4)];
  ((v4f*)part)[((size_t)ich * B_SZ + b) * (DIN / 4) + t] = s;
}

// ---------------------------------------------------------------------------
// Stage 1b: combine the ICH partials -> S[b][k]  (deterministic order, b128).
// 16 blocks x 256 threads, one v4f per thread.
// ---------------------------------------------------------------------------
__global__ void caps_combine(const float* __restrict__ part,
                             float* __restrict__ S) {
  const int idx = blockIdx.x * 256 + threadIdx.x;   // 0..4095 (v4f units)
  const v4f* p = (const v4f*)part;
  v4f s = {};
#pragma unroll
  for (int ich = 0; ich < ICH; ++ich)
    s += p[(size_t)ich * (B_SZ * DIN / 4) + idx];
  ((v4f*)S)[idx] = s;
}

// ---------------------------------------------------------------------------
// Stage 2: O[32x512] = S[32x512] @ W[512x512] with V_WMMA_F32_16X16X4_F32.
// One 16x16 output tile per wave; 2 M-tiles x 32 N-tiles = 64 waves
// = 16 blocks x 4 waves.  K loop: 512/4 = 128 WMMAs per wave.
//
// Per-lane fragment layout (CDNA5 ISA, 32-bit 16x4 A / 4x16 B):
//   lanes 0-15 : K offsets {0,1} of each 4-wide K step, row/col = lane
//   lanes 16-31: K offsets {2,3},                       row/col = lane-16
// C/D 16x16 f32: VGPR j -> M = j (lanes 0-15) / j+8 (lanes 16-31), N = lane%16.
// ---------------------------------------------------------------------------
__global__ void caps_wmma_gemm(const float* __restrict__ S,
                               const float* __restrict__ W,
                               float* __restrict__ O) {
  const int wave = threadIdx.x >> 5;       // wave32
  const int lane = threadIdx.x & 31;
  const int tile = blockIdx.x * 4 + wave;  // 0..63
  const int mt   = tile & 1;               // M tile: 0..1
  const int nt   = tile >> 1;              // N tile: 0..31
  const int hi   = lane >> 4;              // 0: lanes 0-15, 1: lanes 16-31
  const int l16  = lane & 15;
  const int mrow = mt * 16 + l16;          // A row this lane supplies
  const int ncol = nt * 16 + l16;          // B column this lane supplies

  v8f acc = {};
  for (int k0 = 0; k0 < DIN; k0 += 4) {
    const int kb = k0 + hi * 2;            // this lane's K base within the step
    v2f a, bf;
    a.x  = S[(size_t)mrow * DIN + kb];
    a.y  = S[(size_t)mrow * DIN + kb + 1];
    bf.x = W[(size_t)kb * NOUT + ncol];            // lanes sweep N -> coalesced
    bf.y = W[(size_t)(kb + 1) * NOUT + ncol];
    acc = __builtin_amdgcn_wmma_f32_16x16x4_f32(
        /*neg_a=*/false, a, /*neg_b=*/false, bf,
        /*c_mod=*/(short)0, acc, /*reuse_a=*/false, /*reuse_b=*/false);
  }

  const int mbase = mt * 16 + hi * 8;
#pragma unroll
  for (int j = 0; j < 8; ++j)
    O[(size_t)(mbase + j) * NOUT + ncol] = acc[j];
}

// ---------------------------------------------------------------------------
extern "C" void kernel_launch(void* const* d_in, const int* in_sizes, int n_in,
                              void* d_out, int out_size, void* d_ws, size_t ws_size,
                              hipStream_t stream) {
  const float* u = (const float*)d_in[0];   // [32, 1024, 512] f32
  const float* W = (const float*)d_in[1];   // [1, 512, 512]   f32
  float* out = (float*)d_out;               // [32, 32, 16] -> flat [32, 512]

  float* part = (float*)d_ws;                         // ICH * 32 * 512 floats (1 MB)
  float* S    = part + (size_t)ICH * B_SZ * DIN;      // 32 * 512 floats (64 KB)

  caps_reduce_partial<<<dim3(B_SZ, ICH), 128, 0, stream>>>(u, part);
  caps_combine<<<(B_SZ * DIN / 4) / 256, 256, 0, stream>>>(part, S);
  caps_wmma_gemm<<<16, 128, 0, stream>>>(S, W, out);
}